// RUMamba_30064771072283
// MI455X (gfx1250) — compile-verified
//
#include <hip/hip_runtime.h>
#include <math.h>

// ---- problem constants (from reference) ----
#define SQ      16          // SEQ
#define BN      4           // BATCH
#define DM      512         // D_MODEL
#define DST     128         // D_STATE
#define DINNER  2048
#define NH      64          // NHEADS
#define HD      32          // HEADDIM
#define CONVD   2304        // CONV_DIM
#define NPROJ   2368        // used in_proj channels: [2048, 4416)
#define NCLS    2513
#define KFEAT   4096        // HEADDIM * D_STATE
#define M64     64          // S*B = GEMM M

typedef float v2f __attribute__((ext_vector_type(2)));
typedef float v4f __attribute__((ext_vector_type(4)));
typedef float v8f __attribute__((ext_vector_type(8)));

// Issue 4 WMMA 16x16x4 steps from one contiguous 8-float run per lane.
// K remap inside a K=16 block: lane-half hi covers K = 8*hi .. 8*hi+7;
// WMMA u consumes pair (2u, 2u+1) of that run for BOTH A and B, so lane
// K-positions stay matched and the block sums every K exactly once.
__device__ __forceinline__ v8f wmma4_f32(v8f acc, const float* __restrict__ ap,
                                         const float* __restrict__ wp)
{
    v4f a0 = *(const v4f*)(ap);
    v4f a1 = *(const v4f*)(ap + 4);
    v4f w0 = *(const v4f*)(wp);
    v4f w1 = *(const v4f*)(wp + 4);
    v2f al, wl;
    al = __builtin_shufflevector(a0, a0, 0, 1);
    wl = __builtin_shufflevector(w0, w0, 0, 1);
    acc = __builtin_amdgcn_wmma_f32_16x16x4_f32(false, al, false, wl, (short)0, acc, false, false);
    al = __builtin_shufflevector(a0, a0, 2, 3);
    wl = __builtin_shufflevector(w0, w0, 2, 3);
    acc = __builtin_amdgcn_wmma_f32_16x16x4_f32(false, al, false, wl, (short)0, acc, false, false);
    al = __builtin_shufflevector(a1, a1, 0, 1);
    wl = __builtin_shufflevector(w1, w1, 0, 1);
    acc = __builtin_amdgcn_wmma_f32_16x16x4_f32(false, al, false, wl, (short)0, acc, false, false);
    al = __builtin_shufflevector(a1, a1, 2, 3);
    wl = __builtin_shufflevector(w1, w1, 2, 3);
    acc = __builtin_amdgcn_wmma_f32_16x16x4_f32(false, al, false, wl, (short)0, acc, false, false);
    return acc;
}

// =====================================================================
// Kernel A: in-projection GEMM, rows m = s*BN+b (seq-major), only the
// channels [2048,4416) that are consumed downstream. fp32 WMMA 16x16x4.
// Block = 4 waves; wave w owns M-tile w; block owns one 16-wide N-tile,
// so the weight stream is read from HBM exactly once.
// =====================================================================
__global__ void k_inproj(const float* __restrict__ inp,   // (B,S,DM)
                         const float* __restrict__ W,     // (4416, DM)
                         float* __restrict__ proj)        // (M64, NPROJ)
{
    const int lane = threadIdx.x & 31;
    const int wv   = threadIdx.x >> 5;          // M-tile 0..3
    const int idx  = lane & 15;
    const int hi8  = (lane >> 4) * 8;           // 0 or 8 (K run base in block)
    const int m    = wv * 16 + idx;             // GEMM row for A load
    const int b    = m & 3, s = m >> 2;         // m = s*BN + b
    const int nglb = 2048 + blockIdx.x * 16 + idx;

    const float* arow = inp + (size_t)(b * SQ + s) * DM + hi8;
    const float* wrow = W   + (size_t)nglb * DM + hi8;

    v8f acc = {};
    #pragma unroll 2
    for (int k0 = 0; k0 < DM; k0 += 16)
        acc = wmma4_f32(acc, arow + k0, wrow + k0);

    const int mrow = wv * 16 + (lane >> 4) * 8;
    const int ncol = blockIdx.x * 16 + idx;
    #pragma unroll
    for (int r = 0; r < 8; ++r)
        proj[(size_t)(mrow + r) * NPROJ + ncol] = acc[r];
}

// =====================================================================
// Kernel B: elementwise prep.
//  - 4 silu levels of xBC with cumulative (reversed) conv weights
//  - dt = softplus(raw + dt_bias), dA = exp(-dt * exp(A_log))
// =====================================================================
__global__ void k_prep(const float* __restrict__ proj,
                       const float* __restrict__ convw,   // (CONVD,4)
                       const float* __restrict__ convb,   // (CONVD,)
                       const float* __restrict__ dt_bias, // (NH,)
                       const float* __restrict__ A_log,   // (NH,)
                       float* __restrict__ xk,            // (4, M64, CONVD)
                       float* __restrict__ dt,            // (M64, NH)
                       float* __restrict__ dA)            // (M64, NH)
{
    const int NXK = M64 * CONVD;
    int gid = blockIdx.x * blockDim.x + threadIdx.x;
    if (gid < NXK) {
        int m = gid / CONVD, c = gid % CONVD;
        float w0 = convw[c * 4 + 0], w1 = convw[c * 4 + 1];
        float w2 = convw[c * 4 + 2], w3 = convw[c * 4 + 3];
        float cw0 = w3, cw1 = w3 + w2, cw2 = cw1 + w1, cw3 = cw2 + w0;
        float cw[4] = {cw0, cw1, cw2, cw3};
        float raw = proj[(size_t)m * NPROJ + c];
        float bia = convb[c];
        #pragma unroll
        for (int l = 0; l < 4; ++l) {
            float v  = fmaf(raw, cw[l], bia);
            float sv = v / (1.f + __expf(-v));      // silu
            xk[(size_t)l * NXK + gid] = sv;
        }
    } else {
        int id = gid - NXK;
        if (id < M64 * NH) {
            int m = id >> 6, h = id & 63;
            float x = proj[(size_t)m * NPROJ + CONVD + h] + dt_bias[h];
            float d = (x > 20.f) ? x : log1pf(__expf(x));   // softplus
            dt[id] = d;
            dA[id] = __expf(-d * __expf(A_log[h]));
        }
    }
}

__global__ void k_zero(float* __restrict__ p, int n) {
    int gid = blockIdx.x * blockDim.x + threadIdx.x;
    if (gid < n) p[gid] = 0.f;
}

// =====================================================================
// Kernel C: SSM scan + unroll + head-mean.
// One block per (head, b). Lane owns (hd = tid&31, 16 states).
// Scan over s carries h in registers; for each s the (S-s+1)-step
// "unroll" recurrence runs on a copy and is mean-reduced over heads
// with fp32 global atomics into feats[(b*S+s), hd*128+state].
// =====================================================================
__global__ void k_scan(const float* __restrict__ xk,
                       const float* __restrict__ dt,
                       const float* __restrict__ dA,
                       float* __restrict__ feats)         // (M64, KFEAT)
{
    const int b    = blockIdx.x & 3;
    const int head = blockIdx.x >> 2;
    const int hd   = threadIdx.x & 31;
    const int sg   = threadIdx.x >> 5;      // 0..7 state group
    const int st0  = sg * 16;
    const int NXK  = M64 * CONVD;

    float h[16];
    #pragma unroll
    for (int j = 0; j < 16; ++j) h[j] = 0.f;

    for (int s = 0; s < SQ; ++s) {
        const int m = s * BN + b;
        const float dAs = dA[m * NH + head];
        const float dts = dt[m * NH + head];

        float xv[4], Bv[4][16];
        #pragma unroll
        for (int l = 0; l < 4; ++l) {
            const float* base = xk + (size_t)l * NXK + (size_t)m * CONVD;
            xv[l] = base[head * HD + hd];
            #pragma unroll
            for (int j = 0; j < 16; ++j)
                Bv[l][j] = base[DINNER + st0 + j];   // wave-uniform -> scalar loads
        }

        // roll step (level 0 conv weights)
        #pragma unroll
        for (int j = 0; j < 16; ++j)
            h[j] = fmaf(dAs, h[j], dts * Bv[0][j] * xv[0]);

        // unroll: k = 0 .. S-s, conv level min(k,3)
        float g[16];
        #pragma unroll
        for (int j = 0; j < 16; ++j) g[j] = h[j];
        for (int k = 0; k <= SQ - s; ++k) {
            const int l  = (k < 3) ? k : 3;
            const float dx = dts * xv[l];
            #pragma unroll
            for (int j = 0; j < 16; ++j)
                g[j] = fmaf(dAs, g[j], dx * Bv[l][j]);
        }

        float* f = feats + (size_t)(b * SQ + s) * KFEAT + hd * DST + st0;
        #pragma unroll
        for (int j = 0; j < 16; ++j)
            atomicAdd(f + j, g[j] * (1.f / (float)NH));   // mean over heads
    }
}

// =====================================================================
// Kernel D: classifier GEMM, M=64 K=4096 N=2513, fp32 WMMA 16x16x4.
// Block = 4 waves (all M-tiles) per N-tile; tail N-tile clamps loads and
// guards stores (EXEC stays all-1s around every WMMA).
// =====================================================================
__global__ void k_cls(const float* __restrict__ feats,   // (M64, KFEAT)
                      const float* __restrict__ W,       // (NCLS, KFEAT)
                      const float* __restrict__ bias,    // (NCLS,)
                      float* __restrict__ y)             // (M64, NCLS)
{
    const int lane = threadIdx.x & 31;
    const int wv   = threadIdx.x >> 5;
    const int idx  = lane & 15;
    const int hi8  = (lane >> 4) * 8;
    const int n    = blockIdx.x * 16 + idx;
    const int nc   = (n < NCLS) ? n : (NCLS - 1);

    const float* arow = feats + (size_t)(wv * 16 + idx) * KFEAT + hi8;
    const float* wrow = W     + (size_t)nc * KFEAT + hi8;

    v8f acc = {};
    #pragma unroll 2
    for (int k0 = 0; k0 < KFEAT; k0 += 16)
        acc = wmma4_f32(acc, arow + k0, wrow + k0);

    if (n < NCLS) {
        const float bb = bias[n];
        const int mrow = wv * 16 + (lane >> 4) * 8;
        #pragma unroll
        for (int r = 0; r < 8; ++r)
            y[(size_t)(mrow + r) * NCLS + n] = acc[r] + bb;
    }
}

// =====================================================================
extern "C" void kernel_launch(void* const* d_in, const int* in_sizes, int n_in,
                              void* d_out, int out_size, void* d_ws, size_t ws_size,
                              hipStream_t stream)
{
    const float* inputs    = (const float*)d_in[0];
    const float* in_proj_w = (const float*)d_in[1];
    const float* conv_w    = (const float*)d_in[2];
    const float* conv_b    = (const float*)d_in[3];
    const float* dt_bias   = (const float*)d_in[4];
    const float* A_log     = (const float*)d_in[5];
    const float* cls_w     = (const float*)d_in[6];
    const float* cls_b     = (const float*)d_in[7];
    float* y = (float*)d_out;

    // workspace layout (floats): proj | dt | dA | xk[4] | feats  (~4.05 MB)
    float* ws    = (float*)d_ws;
    float* proj  = ws;                              // M64*NPROJ
    float* dtb   = proj + (size_t)M64 * NPROJ;      // M64*NH
    float* dAb   = dtb  + (size_t)M64 * NH;         // M64*NH
    float* xk    = dAb  + (size_t)M64 * NH;         // 4*M64*CONVD
    float* feats = xk   + (size_t)4 * M64 * CONVD;  // M64*KFEAT

    // A: in-projection GEMM (148 N-tiles x 4 waves)
    k_inproj<<<NPROJ / 16, 128, 0, stream>>>(inputs, in_proj_w, proj);

    // B: conv levels + dt/dA
    const int prep_total = M64 * CONVD + M64 * NH;
    k_prep<<<(prep_total + 255) / 256, 256, 0, stream>>>(
        proj, conv_w, conv_b, dt_bias, A_log, xk, dtb, dAb);

    // zero feats accumulator
    k_zero<<<(M64 * KFEAT + 255) / 256, 256, 0, stream>>>(feats, M64 * KFEAT);

    // C: scan + unroll + head mean
    k_scan<<<BN * NH, 256, 0, stream>>>(xk, dtb, dAb, feats);

    // D: classifier GEMM (158 N-tiles x 4 waves)
    k_cls<<<(NCLS + 15) / 16, 128, 0, stream>>>(feats, cls_w, cls_b, y);
}